// MambaGramLayer_60670708023451
// MI455X (gfx1250) — compile-verified
//
#include <hip/hip_runtime.h>

// Problem constants (from reference)
#define D_ 64
#define B_ 16
#define L_ 32000
#define T_ 256              // chunk length
#define C_ 125              // L / T
#define M_ (B_ * C_)        // 2000 chunk rows; x is contiguous as [M_][T_]

typedef float v2f __attribute__((ext_vector_type(2)));
typedef float v4f __attribute__((ext_vector_type(4)));
typedef float v8f __attribute__((ext_vector_type(8)));

__device__ __forceinline__ float softplus_f(float v) {
  // numerically stable softplus
  return fmaxf(v, 0.0f) + log1pf(expf(-fabsf(v)));
}

// ---------------------------------------------------------------------------
// Kernel 1: build transposed GEMM weights Wt[d][k] = a_d^(T-1-k) * b_d (Re/Im)
// a_d = exp(alpha_d + i*omega_d)  =>  a^p = e^{p*alpha} * (cos p*w, sin p*w)
// ---------------------------------------------------------------------------
__global__ __launch_bounds__(T_) void k_buildW(
    const float* __restrict__ omega, const float* __restrict__ raw_alpha,
    const float* __restrict__ brE, const float* __restrict__ biE,
    float* __restrict__ WtRe, float* __restrict__ WtIm) {
  const int d = blockIdx.x;      // 0..63
  const int k = threadIdx.x;     // 0..255
  const float alpha = -softplus_f(raw_alpha[d]);
  const float om    = omega[d];
  const float p     = (float)(T_ - 1 - k);
  const float mag   = expf(p * alpha);
  float s, c;
  sincosf(p * om, &s, &c);
  const float br = brE[d], bi = biE[d];
  WtRe[d * T_ + k] = mag * (c * br - s * bi);
  WtIm[d * T_ + k] = mag * (s * br + c * bi);
}

// ---------------------------------------------------------------------------
// Kernel 2 (WMMA): chunk summaries  E = X * W   (M=2000, K=256, N=64), Re+Im.
// One wave per 16x16 output tile; 4 waves/block share the same A rows.
// f32 WMMA 16x16x4 layouts (ISA 7.12.2):
//   A 16x4 : lanes 0-15 -> M=lane, VGPR0/1 = K0/K1 ; lanes 16-31 -> K2/K3
//   B  4x16: mirrored (N on lane&15, two K per lane in VGPR0/1)
//   C/D    : VGPR v -> M = v + 8*(lane>>4), N = lane&15
// ---------------------------------------------------------------------------
__global__ __launch_bounds__(128) void k_chunk_gemm(
    const float* __restrict__ x, const float* __restrict__ WtRe,
    const float* __restrict__ WtIm,
    float* __restrict__ eRe, float* __restrict__ eIm) {
  const int lane  = threadIdx.x & 31;
  const int wv    = threadIdx.x >> 5;       // ntile 0..3
  const int m0    = blockIdx.x * 16;        // 125 M-tiles
  const int n0    = wv * 16;
  const int mr    = m0 + (lane & 15);
  const int nr    = n0 + (lane & 15);
  const int khalf = (lane >> 4) * 2;        // 0 or 2

  const float* xrow = x    + (size_t)mr * T_ + khalf;
  const float* wre  = WtRe + (size_t)nr * T_ + khalf;
  const float* wim  = WtIm + (size_t)nr * T_ + khalf;

  v8f accRe = {0.f,0.f,0.f,0.f,0.f,0.f,0.f,0.f};
  v8f accIm = {0.f,0.f,0.f,0.f,0.f,0.f,0.f,0.f};
  for (int k = 0; k < T_; k += 4) {
    v2f a  = *(const v2f*)(xrow + k);
    v2f br = *(const v2f*)(wre + k);
    v2f bi = *(const v2f*)(wim + k);
    accRe = __builtin_amdgcn_wmma_f32_16x16x4_f32(
        false, a, false, br, (short)0, accRe, false, false);
    accIm = __builtin_amdgcn_wmma_f32_16x16x4_f32(
        false, a, false, bi, (short)0, accIm, false, false);
  }

  const int nn = n0 + (lane & 15);
  const int mb = m0 + 8 * (lane >> 4);
#pragma unroll
  for (int v = 0; v < 8; ++v) {
    eRe[(size_t)(mb + v) * D_ + nn] = accRe[v];
    eIm[(size_t)(mb + v) * D_ + nn] = accIm[v];
  }
}

// ---------------------------------------------------------------------------
// Kernel 3: exclusive inter-chunk scan of carries.  S_{c} = a^T S_{c-1} + e_c
// carry[b][c][d] = state entering chunk c. 1024 independent threads.
// ---------------------------------------------------------------------------
__global__ __launch_bounds__(256) void k_carry_scan(
    const float* __restrict__ omega, const float* __restrict__ raw_alpha,
    const float* __restrict__ eRe, const float* __restrict__ eIm,
    float* __restrict__ cRe, float* __restrict__ cIm) {
  const int tid = blockIdx.x * 256 + threadIdx.x;   // 0..1023
  const int b = tid >> 6;
  const int d = tid & 63;
  const float alpha = -softplus_f(raw_alpha[d]);
  const float om    = omega[d];
  const float mag   = expf((float)T_ * alpha);
  float s, c;
  sincosf((float)T_ * om, &s, &c);
  const float aTr = mag * c, aTi = mag * s;
  float Sr = 0.f, Si = 0.f;
  for (int ch = 0; ch < C_; ++ch) {
    const size_t idx = (size_t)(b * C_ + ch) * D_ + d;
    cRe[idx] = Sr;
    cIm[idx] = Si;
    const float er = eRe[idx], ei = eIm[idx];
    const float nSr = aTr * Sr - aTi * Si + er;
    const float nSi = aTr * Si + aTi * Sr + ei;
    Sr = nSr; Si = nSi;
  }
}

// ---------------------------------------------------------------------------
// Kernel 4: emit. One wave per chunk; lane owns channels (2l, 2l+1) so each
// step stores one contiguous 16B float4 -> 512B contiguous per wave per t.
// x chunk (1KB/wave) staged via async global->LDS; outputs are non-temporal
// (256MB stream > 192MB L2).
// ---------------------------------------------------------------------------
__global__ __launch_bounds__(256) void k_scan_emit(
    const float* __restrict__ x, const float* __restrict__ omega,
    const float* __restrict__ raw_alpha, const float* __restrict__ brE,
    const float* __restrict__ biE, const float* __restrict__ cRe,
    const float* __restrict__ cIm, float* __restrict__ out) {
  __shared__ float smem[8 * T_];          // 8 waves * 1KB
  const int lane = threadIdx.x & 31;
  const int wv   = threadIdx.x >> 5;
  const int wg   = blockIdx.x * 8 + wv;   // chunk id = b*C_ + c, 0..1999
  const int d0   = 2 * lane;

  // 1) kick off async staging of this wave's 256 x samples into LDS
  float* xs = &smem[wv * T_];
  {
    const float* gsrc = x + (size_t)wg * T_ + lane * 8;   // 32B per lane
    unsigned ldsOff = (unsigned)(size_t)(&xs[lane * 8]);  // low 32b = LDS offset
    asm volatile("global_load_async_to_lds_b128 %0, %1, off"
                 :: "v"(ldsOff), "v"(gsrc) : "memory");
    asm volatile("global_load_async_to_lds_b128 %0, %1, off offset:16"
                 :: "v"(ldsOff), "v"(gsrc) : "memory");
  }

  // 2) per-lane channel constants while the DMA is in flight
  float ar0, ai0, ar1, ai1;
  {
    const float a0 = -softplus_f(raw_alpha[d0]);
    const float a1 = -softplus_f(raw_alpha[d0 + 1]);
    const float m0 = expf(a0), m1 = expf(a1);
    float s0, c0, s1, c1;
    sincosf(omega[d0], &s0, &c0);
    sincosf(omega[d0 + 1], &s1, &c1);
    ar0 = m0 * c0; ai0 = m0 * s0;
    ar1 = m1 * c1; ai1 = m1 * s1;
  }
  const float br0 = brE[d0], bi0 = biE[d0];
  const float br1 = brE[d0 + 1], bi1 = biE[d0 + 1];
  const size_t cidx = (size_t)wg * D_ + d0;
  float h0r = cRe[cidx], h1r = cRe[cidx + 1];
  float h0i = cIm[cidx], h1i = cIm[cidx + 1];

  // 3) wait for the async staging to land in LDS
  asm volatile("s_wait_asynccnt 0x0" ::: "memory");

  const int b  = wg / C_;
  const int t0 = (wg % C_) * T_;
  float* dst = out + ((size_t)b * L_ + t0) * (D_ * 2) + 4 * lane;
#pragma unroll 4
  for (int k = 0; k < T_; ++k) {
    const float xv = xs[k];                     // LDS broadcast
    const float n0r = fmaf(ar0, h0r, fmaf(-ai0, h0i, br0 * xv));
    const float n0i = fmaf(ar0, h0i, fmaf( ai0, h0r, bi0 * xv));
    const float n1r = fmaf(ar1, h1r, fmaf(-ai1, h1i, br1 * xv));
    const float n1i = fmaf(ar1, h1i, fmaf( ai1, h1r, bi1 * xv));
    h0r = n0r; h0i = n0i; h1r = n1r; h1i = n1i;
    v4f o = {h0r, h0i, h1r, h1i};
    __builtin_nontemporal_store(o, (v4f*)dst);  // 16B contiguous per lane
    dst += D_ * 2;
  }
}

// ---------------------------------------------------------------------------
extern "C" void kernel_launch(void* const* d_in, const int* in_sizes, int n_in,
                              void* d_out, int out_size, void* d_ws, size_t ws_size,
                              hipStream_t stream) {
  const float* x         = (const float*)d_in[0];
  const float* omega     = (const float*)d_in[1];
  const float* raw_alpha = (const float*)d_in[2];
  const float* b_real    = (const float*)d_in[3];
  const float* b_imag    = (const float*)d_in[4];
  float* out = (float*)d_out;

  // workspace carving (~2.2 MB total)
  float* ws   = (float*)d_ws;
  float* WtRe = ws; ws += (size_t)D_ * T_;
  float* WtIm = ws; ws += (size_t)D_ * T_;
  float* eRe  = ws; ws += (size_t)M_ * D_;
  float* eIm  = ws; ws += (size_t)M_ * D_;
  float* cRe  = ws; ws += (size_t)M_ * D_;
  float* cIm  = ws; ws += (size_t)M_ * D_;

  k_buildW    <<<D_,     T_,  0, stream>>>(omega, raw_alpha, b_real, b_imag, WtRe, WtIm);
  k_chunk_gemm<<<C_,     128, 0, stream>>>(x, WtRe, WtIm, eRe, eIm);
  k_carry_scan<<<4,      256, 0, stream>>>(omega, raw_alpha, eRe, eIm, cRe, cIm);
  k_scan_emit <<<M_ / 8, 256, 0, stream>>>(x, omega, raw_alpha, b_real, b_imag,
                                           cRe, cIm, out);
}